// SpatialTransform_39891656245250
// MI455X (gfx1250) — compile-verified
//
#include <hip/hip_runtime.h>

// Problem constants (N=1, C=2, D=H=W=160)
#define DD 160
#define HH 160
#define WW 160
#define NVOX (DD * HH * WW)          // 4,096,000 voxels

// Launch shape: block.x = W = 160 (5 wave32s), grid = (H, D).
// w/h/d come directly from thread/block ids -> no div/mod, no bounds check.
__global__ __launch_bounds__(WW) void spatial_transform_trilerp(
    const float* __restrict__ x0,    // channel 0: (D, H, W)
    const float* __restrict__ x1,    // channel 1: (D, H, W)
    const float* __restrict__ flow,  // (D, H, W, 3)
    float* __restrict__ out0,        // channel 0 out
    float* __restrict__ out1)        // channel 1 out
{
    const int w = threadIdx.x;
    const int h = blockIdx.x;
    const int d = blockIdx.y;
    const int idx = (d * HH + h) * WW + w;

    // gfx1250: global_prefetch_b8 — pull the streaming flow read ~96KB ahead.
    __builtin_prefetch(flow + 3ull * (unsigned long long)(idx + 8192), 0, 0);

    // flow is contiguous float3 per voxel -> coalesced b96 load per lane
    const float* f = flow + 3ull * (unsigned long long)idx;
    const float fl0 = f[0];
    const float fl1 = f[1];
    const float fl2 = f[2];

    // g = sample_grid + flow ; sample_grid[...,0]=w, [...,1]=h, [...,2]=d (exact in fp32)
    const float gx = (float)w + fl0;
    const float gy = (float)h + fl1;
    const float gz = (float)d + fl2;

    // Replicate reference fp32 op order: normalize then grid_sample denormalize.
    const float nx = (gx - 0.5f * (float)WW) / (float)WW * 2.0f;
    const float ny = (gy - 0.5f * (float)HH) / (float)HH * 2.0f;
    const float nz = (gz - 0.5f * (float)DD) / (float)DD * 2.0f;
    const float ix = ((nx + 1.0f) * (float)WW - 1.0f) * 0.5f;
    const float iy = ((ny + 1.0f) * (float)HH - 1.0f) * 0.5f;
    const float iz = ((nz + 1.0f) * (float)DD - 1.0f) * 0.5f;

    const float ix0f = floorf(ix);
    const float iy0f = floorf(iy);
    const float iz0f = floorf(iz);
    const float fx = ix - ix0f;
    const float fy = iy - iy0f;
    const float fz = iz - iz0f;
    const int ix0 = (int)ix0f, iy0 = (int)iy0f, iz0 = (int)iz0f;
    const int ix1 = ix0 + 1,   iy1 = iy0 + 1,   iz1 = iz0 + 1;

    // Per-axis weights with validity folded in (w*valid factorizes per axis).
    float wx[2] = { 1.0f - fx, fx };
    float wy[2] = { 1.0f - fy, fy };
    float wz[2] = { 1.0f - fz, fz };
    wx[0] = (ix0 >= 0 && ix0 < WW) ? wx[0] : 0.0f;
    wx[1] = (ix1 >= 0 && ix1 < WW) ? wx[1] : 0.0f;
    wy[0] = (iy0 >= 0 && iy0 < HH) ? wy[0] : 0.0f;
    wy[1] = (iy1 >= 0 && iy1 < HH) ? wy[1] : 0.0f;
    wz[0] = (iz0 >= 0 && iz0 < DD) ? wz[0] : 0.0f;
    wz[1] = (iz1 >= 0 && iz1 < DD) ? wz[1] : 0.0f;

    // Clamped indices so even zero-weight corner loads are in-bounds.
    const int cx[2] = { min(max(ix0, 0), WW - 1), min(max(ix1, 0), WW - 1) };
    const int cy[2] = { min(max(iy0, 0), HH - 1), min(max(iy1, 0), HH - 1) };
    const int cz[2] = { min(max(iz0, 0), DD - 1), min(max(iz1, 0), DD - 1) };

    float acc0 = 0.0f;
    float acc1 = 0.0f;

    #pragma unroll
    for (int a = 0; a < 2; ++a) {
        const int zoff = cz[a] * (HH * WW);
        #pragma unroll
        for (int b = 0; b < 2; ++b) {
            const int yzoff = zoff + cy[b] * WW;
            const float wzy = wz[a] * wy[b];
            #pragma unroll
            for (int c = 0; c < 2; ++c) {
                const float wgt = wzy * wx[c];
                const int off = yzoff + cx[c];
                // Channels share one 32-bit voffset against two SGPR bases;
                // gathers are L2-resident (x = 33MB << 192MB L2).
                acc0 = fmaf(wgt, x0[off], acc0);
                acc1 = fmaf(wgt, x1[off], acc1);
            }
        }
    }

    out0[idx] = acc0;
    out1[idx] = acc1;
}

extern "C" void kernel_launch(void* const* d_in, const int* in_sizes, int n_in,
                              void* d_out, int out_size, void* d_ws, size_t ws_size,
                              hipStream_t stream) {
    (void)in_sizes; (void)n_in; (void)d_ws; (void)ws_size; (void)out_size;
    const float* x    = (const float*)d_in[0];  // (1,2,160,160,160) fp32
    const float* flow = (const float*)d_in[1];  // (1,160,160,160,3) fp32
    // d_in[2] (sample_grid) is an identity meshgrid; recomputed from voxel
    // index in-kernel — saves 49 MB of HBM reads on a bandwidth-bound kernel.
    float* out = (float*)d_out;                 // (1,2,160,160,160) fp32

    dim3 block(WW, 1, 1);                       // 160 = 5 wave32s
    dim3 grid(HH, DD, 1);                       // 160 x 160 blocks
    spatial_transform_trilerp<<<grid, block, 0, stream>>>(
        x, x + NVOX, flow, out, out + NVOX);
}